// RocketFeatures_45054206935560
// MI455X (gfx1250) — compile-verified
//
#include <hip/hip_runtime.h>
#include <stdint.h>
#include <string.h>

typedef __attribute__((ext_vector_type(2))) float v2f;
typedef __attribute__((ext_vector_type(8))) float v8f;

#define N_KERNELS 10000
#define IN_CH 3
#define BATCH 64
#define T_LEN 1024
#define MAXK 11
#define PAD_MAX 80                 // max (k-1)*d/2 = 10*16/2
#define XROW (T_LEN + 2 * PAD_MAX) // 1184 floats per channel in LDS

// ---------------------------------------------------------------------------
// Device kernel: one (k,d) group. One wave = one 16-kernel tile x one batch n.
// Conv expressed as chained V_WMMA_F32_16X16X4_F32 over K = 3k (zero-padded).
// ---------------------------------------------------------------------------
template <int K>
__global__ __launch_bounds__(256) void rocket_group_kernel(
    const float* __restrict__ x,       // [64,3,1024]
    const float* __restrict__ weights, // [10000,3,11]
    const float* __restrict__ biases,  // [10000]
    const int* __restrict__ perm,      // [10000] group-major kernel order
    float* __restrict__ out,           // [64, 20000]
    int g_start, int g_count, int D)
{
    constexpr int KTOT = IN_CH * K;        // 21 / 27 / 33
    constexpr int NKB  = (KTOT + 3) >> 2;  // 6 / 7 / 9 K-blocks of 4

    __shared__ float xs[IN_CH * XROW];

    const int n   = blockIdx.x;  // batch row
    const int tid = threadIdx.x;

    // Stage x[n] into LDS with zeroed "same"-padding halos (shared by 8 waves).
    for (int i = tid; i < IN_CH * XROW; i += 256) xs[i] = 0.0f;
    __syncthreads();
    for (int i = tid; i < IN_CH * T_LEN; i += 256) {
        int c = i >> 10, t = i & (T_LEN - 1);
        xs[c * XROW + PAD_MAX + t] = x[(n * IN_CH + c) * T_LEN + t];
    }
    __syncthreads();

    const int wave   = tid >> 5;
    const int lane   = tid & 31;
    const int tile   = blockIdx.y * 8 + wave;
    const int ntiles = (g_count + 15) >> 4;
    if (tile >= ntiles) return;          // wave-uniform: EXEC stays all-ones

    const int m   = lane & 15;   // A row (kernel) == B column (time-in-tile)
    const int hi  = lane >> 4;   // K sub-pair select: kk = 4i + 2*hi
    const int pad = ((K - 1) * D) >> 1;

    // ---- A tile (16 kernels x KTOT, zero-padded to 4*NKB) + B LDS offsets ----
    const bool rowok = (tile * 16 + m) < g_count;
    const int  o     = rowok ? perm[g_start + tile * 16 + m] : 0;
    const float* wrow = weights + (size_t)o * (IN_CH * MAXK);

    v2f A[NKB];
    int boff0[NKB], boff1[NKB];
#pragma unroll
    for (int i = 0; i < NKB; ++i) {
        const int kk = 4 * i + 2 * hi;
        float a0 = 0.0f, a1 = 0.0f;
        int   b0 = 0, b1 = 0;
        if (kk < KTOT) {
            int c = kk / K, j = kk - c * K;
            if (rowok) a0 = wrow[c * MAXK + j];
            b0 = c * XROW + PAD_MAX + m - pad + j * D;
        }
        if (kk + 1 < KTOT) {
            int c = (kk + 1) / K, j = (kk + 1) - c * K;
            if (rowok) a1 = wrow[c * MAXK + j];
            b1 = c * XROW + PAD_MAX + m - pad + j * D;
        }
        A[i].x = a0; A[i].y = a1;
        boff0[i] = b0; boff1[i] = b1;
    }

    // Bias per accumulator row (VGPR r holds kernel row r + 8*hi).
    float bias_v[8];
#pragma unroll
    for (int r = 0; r < 8; ++r) {
        int mm = tile * 16 + r + 8 * hi;
        bias_v[r] = (mm < g_count) ? biases[perm[g_start + mm]] : 0.0f;
    }

    float vmax[8], vcnt[8];
#pragma unroll
    for (int r = 0; r < 8; ++r) { vmax[r] = -3.402823466e38f; vcnt[r] = 0.0f; }

    // ---- main loop: 64 time tiles of 16, NKB chained fp32 WMMAs each ----
    for (int tb = 0; tb < T_LEN; tb += 16) {
        v8f acc = {};
#pragma unroll
        for (int i = 0; i < NKB; ++i) {
            v2f b;
            b.x = xs[boff0[i] + tb];
            b.y = xs[boff1[i] + tb];
            acc = __builtin_amdgcn_wmma_f32_16x16x4_f32(
                false, A[i], false, b, (short)0, acc, false, false);
        }
#pragma unroll
        for (int r = 0; r < 8; ++r) {
            float v = acc[r] + bias_v[r];
            vmax[r] = fmaxf(vmax[r], v);
            vcnt[r] += (v > 0.0f) ? 1.0f : 0.0f;
        }
    }

    // ---- reduce across the 16 time lanes in each half-wave, scatter out ----
#pragma unroll
    for (int r = 0; r < 8; ++r) {
        float mx = vmax[r], ct = vcnt[r];
#pragma unroll
        for (int s = 8; s >= 1; s >>= 1) {
            mx = fmaxf(mx, __shfl_xor(mx, s, 32));
            ct += __shfl_xor(ct, s, 32);
        }
        if (m == 0) {
            int mm = tile * 16 + r + 8 * hi;
            if (mm < g_count) {
                int orig = perm[g_start + mm];
                size_t base = (size_t)n * (2 * N_KERNELS) + 2 * (size_t)orig;
                out[base + 0] = mx;
                out[base + 1] = ct * (1.0f / T_LEN);
            }
        }
    }
}

// ---------------------------------------------------------------------------
// Host: replicate numpy default_rng(0) (SeedSequence -> PCG64 -> Lemire
// bounded integers) to recover the per-kernel (k, d) assignment, then build
// the group-major permutation exactly like the reference.
// ---------------------------------------------------------------------------
static int g_perm[N_KERNELS];
static int g_ks[N_KERNELS], g_dl[N_KERNELS];

static void compute_groups(int* gs, int* gc, int* gk, int* gd, int* ng_out)
{
    typedef unsigned __int128 u128;
    const uint32_t INIT_A = 0x43b0d7e5u, MULT_A = 0x931e8875u;
    const uint32_t INIT_B = 0x8b51f9ddu, MULT_B = 0x58f38dedu;
    const uint32_t MIX_L = 0xca01f9ddu, MIX_R = 0x4973f715u;

    // SeedSequence(0), pool_size=4
    uint32_t pool[4];
    uint32_t hc = INIT_A;
    auto hashmix = [&](uint32_t v) -> uint32_t {
        v ^= hc; hc *= MULT_A; v *= hc; v ^= v >> 16; return v;
    };
    auto mix = [&](uint32_t xx, uint32_t yy) -> uint32_t {
        uint32_t r = xx * MIX_L - yy * MIX_R; r ^= r >> 16; return r;
    };
    for (int i = 0; i < 4; ++i) pool[i] = hashmix(i == 0 ? 0u : 0u);
    for (int s = 0; s < 4; ++s)
        for (int d = 0; d < 4; ++d)
            if (s != d) pool[d] = mix(pool[d], hashmix(pool[s]));

    // generate_state(4, uint64) -> 8 uint32 words
    uint32_t st[8];
    uint32_t hb = INIT_B;
    for (int i = 0; i < 8; ++i) {
        uint32_t v = pool[i & 3];
        v ^= hb; hb *= MULT_B; v *= hb; v ^= v >> 16;
        st[i] = v;
    }
    uint64_t v64[4];
    for (int i = 0; i < 4; ++i)
        v64[i] = (uint64_t)st[2 * i] | ((uint64_t)st[2 * i + 1] << 32);

    // PCG64 init: seed = (v64[0]<<64)|v64[1], inc-seq = (v64[2]<<64)|v64[3]
    const u128 MULT = ((u128)0x2360ed051fc65da4ULL << 64) | 0x4385df649fccf645ULL;
    u128 seed = ((u128)v64[0] << 64) | v64[1];
    u128 iseq = ((u128)v64[2] << 64) | v64[3];
    u128 inc = (iseq << 1) | 1;
    u128 state = 0;
    state = state * MULT + inc;
    state += seed;
    state = state * MULT + inc;

    bool has32 = false;
    uint32_t saved32 = 0;
    auto next64 = [&]() -> uint64_t {
        state = state * MULT + inc;
        uint64_t hi = (uint64_t)(state >> 64), lo = (uint64_t)state;
        unsigned rot = (unsigned)(state >> 122);
        uint64_t xr = hi ^ lo;
        return (xr >> rot) | (xr << ((64u - rot) & 63u));
    };
    auto next32 = [&]() -> uint32_t {
        if (has32) { has32 = false; return saved32; }
        uint64_t n = next64();
        has32 = true; saved32 = (uint32_t)(n >> 32);
        return (uint32_t)n;
    };
    auto lemire = [&](uint32_t rng_excl) -> uint32_t { // uniform in [0, rng_excl)
        uint32_t rng = rng_excl - 1;
        uint64_t mv = (uint64_t)next32() * (uint64_t)rng_excl;
        uint32_t leftover = (uint32_t)mv;
        if (leftover < rng_excl) {
            uint32_t threshold = (0xFFFFFFFFu - rng) % rng_excl;
            while (leftover < threshold) {
                mv = (uint64_t)next32() * (uint64_t)rng_excl;
                leftover = (uint32_t)mv;
            }
        }
        return (uint32_t)(mv >> 32);
    };

    static const int KS[3] = {7, 9, 11};
    static const int DL[5] = {1, 2, 4, 8, 16};
    for (int i = 0; i < N_KERNELS; ++i) g_ks[i] = KS[lemire(3)];
    for (int i = 0; i < N_KERNELS; ++i) g_dl[i] = DL[lemire(5)];

    int ng = 0, pos = 0;
    for (int ki = 0; ki < 3; ++ki) {
        for (int di = 0; di < 5; ++di) {
            int startp = pos, cnt = 0;
            for (int i = 0; i < N_KERNELS; ++i)
                if (g_ks[i] == KS[ki] && g_dl[i] == DL[di]) { g_perm[pos++] = i; ++cnt; }
            if (cnt) { gs[ng] = startp; gc[ng] = cnt; gk[ng] = KS[ki]; gd[ng] = DL[di]; ++ng; }
        }
    }
    *ng_out = ng;
}

extern "C" void kernel_launch(void* const* d_in, const int* in_sizes, int n_in,
                              void* d_out, int out_size, void* d_ws, size_t ws_size,
                              hipStream_t stream)
{
    (void)in_sizes; (void)n_in; (void)out_size; (void)ws_size;
    const float* x = (const float*)d_in[0];
    const float* w = (const float*)d_in[1];
    const float* b = (const float*)d_in[2];
    float* out = (float*)d_out;

    static int gs[16], gc[16], gk[16], gd[16];
    int ng = 0;
    compute_groups(gs, gc, gk, gd, &ng);

    int* d_perm = (int*)d_ws;
    hipMemcpyAsync(d_perm, g_perm, N_KERNELS * sizeof(int),
                   hipMemcpyHostToDevice, stream);

    for (int g = 0; g < ng; ++g) {
        int ntiles = (gc[g] + 15) / 16;
        dim3 grid(BATCH, (ntiles + 7) / 8);
        switch (gk[g]) {
        case 7:
            rocket_group_kernel<7><<<grid, 256, 0, stream>>>(x, w, b, d_perm, out, gs[g], gc[g], gd[g]);
            break;
        case 9:
            rocket_group_kernel<9><<<grid, 256, 0, stream>>>(x, w, b, d_perm, out, gs[g], gc[g], gd[g]);
            break;
        default:
            rocket_group_kernel<11><<<grid, 256, 0, stream>>>(x, w, b, d_perm, out, gs[g], gc[g], gd[g]);
            break;
        }
    }
}